// SpecCnn1D_32358283608249
// MI455X (gfx1250) — compile-verified
//
#include <hip/hip_runtime.h>

typedef float v2f __attribute__((ext_vector_type(2)));
typedef float v8f __attribute__((ext_vector_type(8)));

#define F_  64
#define K_  5
#define L_  4096
#define B_  256
#define PAD_ 2
#define I_  (L_ + 2 * PAD_)   // 4100
#define O_  L_                // 4096
#define OC  16                // o positions per block chunk
#define XS_W (OC + 4)         // staged window width per batch row

// Each wave: 16(batch) x 16(feature) output tile at position o.
// y[b,o,f] = sum_k x_pad[b,o+k] * phi[f,k] * (lam_in[f,o+k] - lam_out[f,o]) + bias[f], relu.
// K=5 reduction done with two chained V_WMMA_F32_16X16X4_F32 (K padded 5->8 with zeros).
__global__ __launch_bounds__(128) void speccnn_wmma_f32(
    const float* __restrict__ x,        // (B, L)
    const float* __restrict__ phi,      // (F, K)
    const float* __restrict__ lam_in,   // (F, I)
    const float* __restrict__ lam_out,  // (F, O)
    const float* __restrict__ bias,     // (F,)
    float* __restrict__ out)            // (B, O, F)
{
    __shared__ float xs[16 * XS_W];

    const int b0     = blockIdx.x * 16;   // batch tile
    const int o_base = blockIdx.y * OC;   // o chunk
    const int t      = threadIdx.x;

    // Stage padded x window for 16 batch rows into LDS (zero-padded edges).
    for (int idx = t; idx < 16 * XS_W; idx += 128) {
        const int i = idx / XS_W;         // batch row in tile
        const int j = idx % XS_W;         // window column
        const int p = o_base + j;         // x_pad position
        float v = 0.0f;
        if (p >= PAD_ && p < L_ + PAD_)
            v = x[(size_t)(b0 + i) * L_ + (p - PAD_)];
        xs[idx] = v;
    }
    __syncthreads();

    const int lane = t & 31;
    const int wave = t >> 5;              // 0..3 -> feature tile
    const int m    = lane & 15;           // A: row M; B/C: col N
    const int half = lane >> 4;           // 0: K=0/1 (A/B v0/v1), 1: K=2/3
    const int f    = wave * 16 + m;

    // Loop-invariant per-lane weights.
    const int   kA0   = half ? 2 : 0;
    const int   kA1   = kA0 + 1;
    const float amask = half ? 0.0f : 1.0f;           // kills K=6 pad slot without a branch
    const float p0    = phi[f * K_ + kA0];
    const float p1    = phi[f * K_ + kA1];
    const float p4    = half ? 0.0f : phi[f * K_ + 4]; // second WMMA: K=4 live, K=5/6/7 pad
    const float bf    = bias[f];
    const float* __restrict__ li   = lam_in  + (size_t)f * I_;
    const float* __restrict__ lo   = lam_out + (size_t)f * O_;
    const float* __restrict__ xrow = xs + m * XS_W;

    const size_t bstride = (size_t)O_ * F_;   // out stride per batch row

    for (int oi = 0; oi < OC; ++oi) {
        const int o   = o_base + oi;
        const float lov = lo[o];

        // A matrix (x window) — ISA 16x4 f32 layout; pad slots zeroed via mask mul.
        v2f A1, A2;
        A1.x = xrow[oi + kA0];
        A1.y = xrow[oi + kA1];
        A2.x = amask * xrow[oi + 4];          // K=4 (lanes 0-15); K=6 pad -> 0
        A2.y = 0.0f;                          // K=5 / K=7 pad

        // B matrix (per-position combined weight) — mirrored 4x16 layout.
        v2f Bm1, Bm2;
        Bm1.x = p0 * (li[o + kA0] - lov);
        Bm1.y = p1 * (li[o + kA1] - lov);
        Bm2.x = p4 * (li[o + 4] - lov);       // zero for half==1 (p4==0)
        Bm2.y = 0.0f;

        v8f c = {};
        c = __builtin_amdgcn_wmma_f32_16x16x4_f32(false, A1, false, Bm1, (short)0, c, false, false);
        c = __builtin_amdgcn_wmma_f32_16x16x4_f32(false, A2, false, Bm2, (short)0, c, false, false);

        // Epilogue: bias + relu, coalesced non-temporal stores
        // (lanes 0-15 contiguous 64B segment, lanes 16-31 ditto on another batch row).
        float* op = out + ((size_t)(b0 + 8 * half) * O_ + o) * F_ + wave * 16 + m;
#pragma unroll
        for (int r = 0; r < 8; ++r) {
            float y = c[r] + bf;
            y = y > 0.0f ? y : 0.0f;
            __builtin_nontemporal_store(y, op + (size_t)r * bstride);
        }
    }
}

extern "C" void kernel_launch(void* const* d_in, const int* in_sizes, int n_in,
                              void* d_out, int out_size, void* d_ws, size_t ws_size,
                              hipStream_t stream) {
    const float* x       = (const float*)d_in[0];
    const float* phi     = (const float*)d_in[1];
    const float* lam_in  = (const float*)d_in[2];
    const float* lam_out = (const float*)d_in[3];
    const float* bias    = (const float*)d_in[4];
    float* out = (float*)d_out;

    dim3 grid(B_ / 16, O_ / OC);   // 16 batch tiles x 256 o-chunks
    speccnn_wmma_f32<<<grid, 128, 0, stream>>>(x, phi, lam_in, lam_out, bias, out);
}